// RoIPooling2DTorch_74148315398540
// MI455X (gfx1250) — compile-verified
//
#include <hip/hip_runtime.h>
#include <hip/hip_bf16.h>

// Problem constants (from reference setup_inputs)
#define RR 256   // number of ROIs
#define CH 256   // channels
#define HH 64    // feature height
#define WW 64    // feature width
#define PP 7     // pooled output 7x7
#define TC 4     // channels staged per block
#define NBLK_C (CH / TC)   // 64 channel-blocks
#define MAXROWS 63         // Lh = x1-x0 <= 63 after clamping to [0,63]

__device__ __forceinline__ int iclamp(int v, int lo, int hi) {
  return v < lo ? lo : (v > hi ? hi : v);
}

__global__ __launch_bounds__(256) void roi_pool_cdna5_kernel(
    const float* __restrict__ features,   // [4,256,64,64]
    const float* __restrict__ rois,       // [256,5]
    float* __restrict__ out) {            // [256,256,7,7]
  // 4 channels x up to 63 rows x 64 cols (full-width layout; only the
  // segments covering [y0,y1) are filled by the async DMA).
  __shared__ float tile[TC][MAXROWS * WW];   // 63 KB static LDS

  const int bx  = blockIdx.x;
  const int r   = bx / NBLK_C;
  const int cb  = bx % NBLK_C;
  const int c0  = cb * TC;
  const int tid = threadIdx.x;

  // ---- Uniform per-block ROI box math (round half-to-even like np.round).
  const float* roi = rois + r * 5;
  const float scale = 1.0f / 32.0f;
  const int b   = (int)roi[0];
  const int bx0 = (int)rintf(roi[1] * scale);   // s_rndne_f32: half-to-even
  const int by0 = (int)rintf(roi[2] * scale);
  const int bx1 = (int)rintf(roi[3] * scale);
  const int by1 = (int)rintf(roi[4] * scale);
  const int x0 = iclamp(bx0, 0, WW - 1);
  const int y0 = iclamp(by0, 0, HH - 1);
  const int x1 = iclamp(bx1, 0, WW - 1);
  const int y1 = iclamp(by1, 0, HH - 1);
  const bool valid = (x0 < x1) && (y0 < y1);

  if (!valid) {
    // Reference leaves degenerate ROIs at zero; d_out is poisoned, so write it.
    if (tid < TC * PP * PP) {
      const int cc  = tid / (PP * PP);
      const int bin = tid % (PP * PP);
      __builtin_nontemporal_store(
          0.0f,
          &out[((size_t)(r * CH + c0 + cc) * PP + bin / PP) * PP + (bin % PP)]);
    }
    return;
  }

  const int Lh = x1 - x0;   // H-axis window [x0, x1)  (x range -> rows, per reference quirk)
  const int Lw = y1 - y0;   // W-axis window [y0, y1)  (y range -> cols)

  // ---- Stage only the 16B-aligned column segments covering [y0, y1),
  // rows [x0, x1), via CDNA5 async LDS DMA (B128 chunks, ASYNCcnt-tracked).
  const unsigned long long gbase = (unsigned long long)features;
  const int seg_lo = y0 >> 2;             // first float4 segment needed
  const int seg_hi = (y1 + 3) >> 2;       // one past last segment needed
  const int nseg   = seg_hi - seg_lo;     // 1..16 segments per row

  const int lane_seg = tid & 15;          // which segment this lane copies
  const int row_phase = tid >> 4;         // 16 rows in flight across the block
  if (lane_seg < nseg) {
    const int col4 = (seg_lo + lane_seg) * 4;   // absolute starting column
#pragma unroll
    for (int cc = 0; cc < TC; ++cc) {
      const int chan_base = ((b * CH + (c0 + cc)) * HH) * WW;
      for (int rrow = row_phase; rrow < Lh; rrow += 16) {
        const int row = x0 + rrow;
        // LDS byte address: generic shared address truncated to low 32 bits.
        unsigned lds_off = (unsigned)(size_t)&tile[cc][rrow * WW + col4];
        // Global byte offset from SGPR base (max ~16.8 MB, fits signed 32-bit).
        unsigned goff = (unsigned)((chan_base + row * WW + col4) * (int)sizeof(float));
        // GVS mode: dsaddr VGPR, 32-bit voffset VGPR, 64-bit SGPR base.
        asm volatile("global_load_async_to_lds_b128 %0, %1, %2 offset:0"
                     :
                     : "v"(lds_off), "v"(goff), "s"(gbase)
                     : "memory");
      }
    }
  }

  // Drain this wave's ASYNCcnt, then make all waves' LDS writes visible.
#if __has_builtin(__builtin_amdgcn_s_wait_asynccnt)
  __builtin_amdgcn_s_wait_asynccnt(0);
#else
  asm volatile("s_wait_asynccnt 0x0" ::: "memory");
#endif
  __syncthreads();

  // ---- Each of threads 0..195 owns one (channel, ph, pw) output bin.
  if (tid < TC * PP * PP) {
    const int cc  = tid / (PP * PP);
    const int bin = tid % (PP * PP);
    const int ph  = bin / PP;
    const int pw  = bin % PP;
    // adaptive_max_pool2d bin bounds: [floor(i*L/P), ceil((i+1)*L/P))
    const int hs = (ph * Lh) / PP;                       // tile-relative row
    const int he = ((ph + 1) * Lh + (PP - 1)) / PP;
    const int ws = y0 + (pw * Lw) / PP;                  // absolute col
    const int we = y0 + ((pw + 1) * Lw + (PP - 1)) / PP;
    float m = -INFINITY;
    for (int h = hs; h < he; ++h) {
      const float* rowp = &tile[cc][h * WW];
      for (int w = ws; w < we; ++w) {
        m = fmaxf(m, rowp[w]);
      }
    }
    __builtin_nontemporal_store(
        m, &out[((size_t)(r * CH + c0 + cc) * PP + ph) * PP + pw]);
  }
}

extern "C" void kernel_launch(void* const* d_in, const int* in_sizes, int n_in,
                              void* d_out, int out_size, void* d_ws, size_t ws_size,
                              hipStream_t stream) {
  (void)in_sizes; (void)n_in; (void)out_size; (void)d_ws; (void)ws_size;
  const float* features = (const float*)d_in[0];   // [4,256,64,64] f32
  const float* rois     = (const float*)d_in[1];   // [256,5] f32
  float* out            = (float*)d_out;           // [256,256,7,7] f32

  dim3 grid(RR * NBLK_C);   // 16384 blocks: (roi, 4-channel slab)
  dim3 block(256);          // 8 wave32s
  roi_pool_cdna5_kernel<<<grid, block, 0, stream>>>(features, rois, out);
}